// LitEma_42657615184219
// MI455X (gfx1250) — compile-verified
//
#include <hip/hip_runtime.h>

// LitEma fused EMA update: out = shadow - (1-decay)*(shadow - params)
// Memory-bound streaming kernel: 12 B/elem, 768 MiB total, ~34.5 us floor at
// 23.3 TB/s HBM. Strategy: b128 non-temporal loads/stores (stream >> 192MB L2),
// 16 elems/thread (8 b128 loads in flight per wave before first wait),
// 32-bit offsets off 64-bit base (GVS addressing), 256-thread wave32 blocks.

typedef float v4f __attribute__((ext_vector_type(4)));

#define EMA_DECAY 0.9999f
#define ELEMS_PER_THREAD 16  // four float4 per input stream

__global__ __launch_bounds__(256) void litema_ema_kernel(
    const float* __restrict__ shadow,
    const float* __restrict__ params,
    const int*   __restrict__ num_updates,
    float*       __restrict__ out,
    int n)  // n = 67108864 fits in int; byte offsets < 2^31
{
    // Uniform scalar: decay_val = min(DECAY, (1+nu)/(10+nu)), nu = num_updates+1.
    // Exact IEEE divide (v_div_scale/v_div_fmas chain) to match reference; runs
    // once per thread, negligible vs. memory traffic.
    float nu    = (float)(num_updates[0] + 1);
    float decay = fminf(EMA_DECAY, (1.0f + nu) / (10.0f + nu));
    float omd   = 1.0f - decay;  // one_minus_decay

    int base = ((int)blockIdx.x * (int)blockDim.x + (int)threadIdx.x) * ELEMS_PER_THREAD;

    if (base + ELEMS_PER_THREAD <= n) {
        const v4f* s = (const v4f*)(shadow + base);
        const v4f* p = (const v4f*)(params + base);
        // Issue all 8 b128 NT loads before any use -> 8 loads in flight per lane
        v4f s0 = __builtin_nontemporal_load(s + 0);
        v4f s1 = __builtin_nontemporal_load(s + 1);
        v4f s2 = __builtin_nontemporal_load(s + 2);
        v4f s3 = __builtin_nontemporal_load(s + 3);
        v4f p0 = __builtin_nontemporal_load(p + 0);
        v4f p1 = __builtin_nontemporal_load(p + 1);
        v4f p2 = __builtin_nontemporal_load(p + 2);
        v4f p3 = __builtin_nontemporal_load(p + 3);

        // shadow - omd*(shadow - params), same op order as reference (FMA-fused)
        v4f r0 = s0 - omd * (s0 - p0);
        v4f r1 = s1 - omd * (s1 - p1);
        v4f r2 = s2 - omd * (s2 - p2);
        v4f r3 = s3 - omd * (s3 - p3);

        v4f* o = (v4f*)(out + base);
        __builtin_nontemporal_store(r0, o + 0);
        __builtin_nontemporal_store(r1, o + 1);
        __builtin_nontemporal_store(r2, o + 2);
        __builtin_nontemporal_store(r3, o + 3);
    } else {
        // Tail (never taken for N=64Mi = 16*4Mi, kept for safety)
        for (int i = base; i < n; ++i) {
            float sv = shadow[i];
            float pv = params[i];
            out[i] = sv - omd * (sv - pv);
        }
    }
}

extern "C" void kernel_launch(void* const* d_in, const int* in_sizes, int n_in,
                              void* d_out, int out_size, void* d_ws, size_t ws_size,
                              hipStream_t stream) {
    const float* shadow      = (const float*)d_in[0];
    const float* params      = (const float*)d_in[1];
    const int*   num_updates = (const int*)d_in[2];
    float*       out         = (float*)d_out;

    int n = in_sizes[0];  // 67108864

    int threads_needed = (n + ELEMS_PER_THREAD - 1) / ELEMS_PER_THREAD;  // 4Mi
    int block = 256;  // 8 wave32 waves per block
    int grid  = (threads_needed + block - 1) / block;                    // 16384

    litema_ema_kernel<<<grid, block, 0, stream>>>(shadow, params,
                                                  num_updates, out, n);
}